// SAGE_34986803593569
// MI455X (gfx1250) — compile-verified
//
#include <hip/hip_runtime.h>
#include <hip/hip_bf16.h>

typedef __attribute__((ext_vector_type(2))) float v2f;
typedef __attribute__((ext_vector_type(8))) float v8f;

// ---------------------------------------------------------------------------
// Zero-fill f32 buffer
// ---------------------------------------------------------------------------
__global__ __launch_bounds__(256) void zero_f32(float* __restrict__ p, unsigned long long n) {
    unsigned long long i = (unsigned long long)blockIdx.x * blockDim.x + threadIdx.x;
    if (i < n) p[i] = 0.0f;
}

// ---------------------------------------------------------------------------
// Edge scatter: one wave32 per edge. Lanes 0..31 each move 4 consecutive
// features (float4 = 16B -> 512B coalesced gather per edge), then 4
// global_atomic_add_f32 into msg[dst]. Lane 0 bumps deg[dst].
// d_in is fixed at 128 for all three layers.
// ---------------------------------------------------------------------------
__global__ __launch_bounds__(256) void sage_scatter(const float* __restrict__ hsrc,
                                                    const int* __restrict__ src,
                                                    const int* __restrict__ dst,
                                                    int nedges,
                                                    float* __restrict__ msg,
                                                    float* __restrict__ deg) {
    int lane = threadIdx.x & 31;
    int edge = blockIdx.x * (blockDim.x >> 5) + (threadIdx.x >> 5);
    if (edge >= nedges) return;
    int s = src[edge];
    int d = dst[edge];
    const float4 v = *(const float4*)(hsrc + (size_t)s * 128 + lane * 4);
    float* mrow = msg + (size_t)d * 128 + lane * 4;
    atomicAdd(mrow + 0, v.x);
    atomicAdd(mrow + 1, v.y);
    atomicAdd(mrow + 2, v.z);
    atomicAdd(mrow + 3, v.w);
    if (lane == 0) atomicAdd(deg + d, 1.0f);
}

// ---------------------------------------------------------------------------
// Fused SAGE layer GEMM via V_WMMA_F32_16X16X4_F32 (full f32 precision):
//   out = [relu]( hdst @ Wself^T + (msg/max(deg,1)) @ Wneigh^T + b )
// One wave per 16x16 output tile; K-loop in steps of 4 over d_in=128.
// f32 WMMA VGPR layout (ISA 7.12.2):
//   A 16x4 : lane(idx,half): a = A[idx][k + 2*half + {0,1}]
//   B 4x16 : lane(idx,half): b = B[k + 2*half + {0,1}][idx] = W[idx][k+2*half+{0,1}]
//   C 16x16: c[j] = D[j + 8*half][idx]
// ---------------------------------------------------------------------------
__global__ __launch_bounds__(256) void sage_gemm(const float* __restrict__ hdst,
                                                 const float* __restrict__ msg,
                                                 const float* __restrict__ deg,
                                                 const float* __restrict__ Wself,
                                                 const float* __restrict__ Wneigh,
                                                 const float* __restrict__ bias,
                                                 float* __restrict__ out,
                                                 int n_dst, int d_out, int relu) {
    const int d_in = 128;
    int lane   = threadIdx.x & 31;
    int wave   = blockIdx.x * (blockDim.x >> 5) + (threadIdx.x >> 5);
    int ntiles = d_out >> 4;
    int mtiles = (n_dst + 15) >> 4;
    if (wave >= mtiles * ntiles) return;

    int mt = wave / ntiles;
    int nt = wave - mt * ntiles;
    int m0 = mt << 4;
    int n0 = nt << 4;

    int half = lane >> 4;
    int idx  = lane & 15;

    // Clamp A-row for the ragged last M-tile (stores are masked below);
    // EXEC stays all-ones as WMMA requires.
    int arow = m0 + idx;
    if (arow > n_dst - 1) arow = n_dst - 1;

    const float* Ah = hdst  + (size_t)arow * d_in + 2 * half;
    const float* Am = msg   + (size_t)arow * d_in + 2 * half;
    const float* Bs = Wself  + (size_t)(n0 + idx) * d_in + 2 * half;
    const float* Bn = Wneigh + (size_t)(n0 + idx) * d_in + 2 * half;
    float invdeg = 1.0f / fmaxf(deg[arow], 1.0f);

    v8f c = {0.f, 0.f, 0.f, 0.f, 0.f, 0.f, 0.f, 0.f};

    // Self GEMM: c += hdst_tile @ Wself^T_tile
#pragma unroll
    for (int k = 0; k < d_in; k += 4) {
        v2f a = *(const v2f*)(Ah + k);
        v2f b = *(const v2f*)(Bs + k);
        c = __builtin_amdgcn_wmma_f32_16x16x4_f32(false, a, false, b,
                                                  (short)0, c, false, false);
    }
    // Neighbor GEMM: c += (msg_tile * invdeg_row) @ Wneigh^T_tile
#pragma unroll
    for (int k = 0; k < d_in; k += 4) {
        v2f a = *(const v2f*)(Am + k);
        a.x *= invdeg;
        a.y *= invdeg;
        v2f b = *(const v2f*)(Bn + k);
        c = __builtin_amdgcn_wmma_f32_16x16x4_f32(false, a, false, b,
                                                  (short)0, c, false, false);
    }

    float bv = bias[n0 + idx];
#pragma unroll
    for (int j = 0; j < 8; ++j) {
        int m = m0 + j + 8 * half;
        if (m < n_dst) {
            float v = c[j] + bv;
            if (relu) v = fmaxf(v, 0.0f);
            out[(size_t)m * d_out + (n0 + idx)] = v;
        }
    }
}

// ---------------------------------------------------------------------------
// Launcher
// ---------------------------------------------------------------------------
static inline unsigned cdiv(unsigned a, unsigned b) { return (a + b - 1) / b; }

extern "C" void kernel_launch(void* const* d_in, const int* in_sizes, int n_in,
                              void* d_out, int out_size, void* d_ws, size_t ws_size,
                              hipStream_t stream) {
    const float* feats = (const float*)d_in[0];
    const int* src0 = (const int*)d_in[1];
    const int* dst0 = (const int*)d_in[2];
    const int* src1 = (const int*)d_in[3];
    const int* dst1 = (const int*)d_in[4];
    const int* src2 = (const int*)d_in[5];
    const int* dst2 = (const int*)d_in[6];
    const float* Ws0 = (const float*)d_in[10];
    const float* Wn0 = (const float*)d_in[11];
    const float* b0  = (const float*)d_in[12];
    const float* Ws1 = (const float*)d_in[13];
    const float* Wn1 = (const float*)d_in[14];
    const float* b1  = (const float*)d_in[15];
    const float* Ws2 = (const float*)d_in[16];
    const float* Wn2 = (const float*)d_in[17];
    const float* b2  = (const float*)d_in[18];

    const int E0 = in_sizes[1], E1 = in_sizes[3], E2 = in_sizes[5];
    // Destination node counts (Python scalars, fixed by setup_inputs).
    const int N1 = 100000, N2 = 20000, N3 = 5000;

    // Workspace layout (16B-aligned slices): h1 | h2 | msg (reused) | deg (reused)
    char* ws = (char*)d_ws;
    float* h1  = (float*)ws;                    ws += (size_t)N1 * 128 * sizeof(float);
    float* h2  = (float*)ws;                    ws += (size_t)N2 * 128 * sizeof(float);
    float* msg = (float*)ws;                    ws += (size_t)N1 * 128 * sizeof(float);
    float* deg = (float*)ws;

    const dim3 blk(256);

    // ---------------- Layer 0: feats(500K x 128) -> h1(100K x 128), relu
    zero_f32<<<cdiv((unsigned)N1 * 128, 256), blk, 0, stream>>>(msg, (unsigned long long)N1 * 128);
    zero_f32<<<cdiv((unsigned)N1, 256), blk, 0, stream>>>(deg, (unsigned long long)N1);
    sage_scatter<<<cdiv((unsigned)E0, 8), blk, 0, stream>>>(feats, src0, dst0, E0, msg, deg);
    {
        unsigned tiles = cdiv((unsigned)N1, 16) * (128 / 16);
        sage_gemm<<<cdiv(tiles, 8), blk, 0, stream>>>(feats, msg, deg, Ws0, Wn0, b0, h1, N1, 128, 1);
    }

    // ---------------- Layer 1: h1(100K x 128) -> h2(20K x 128), relu
    zero_f32<<<cdiv((unsigned)N2 * 128, 256), blk, 0, stream>>>(msg, (unsigned long long)N2 * 128);
    zero_f32<<<cdiv((unsigned)N2, 256), blk, 0, stream>>>(deg, (unsigned long long)N2);
    sage_scatter<<<cdiv((unsigned)E1, 8), blk, 0, stream>>>(h1, src1, dst1, E1, msg, deg);
    {
        unsigned tiles = cdiv((unsigned)N2, 16) * (128 / 16);
        sage_gemm<<<cdiv(tiles, 8), blk, 0, stream>>>(h1, msg, deg, Ws1, Wn1, b1, h2, N2, 128, 1);
    }

    // ---------------- Layer 2: h2(20K x 128) -> out(5K x 64), no relu
    zero_f32<<<cdiv((unsigned)N3 * 128, 256), blk, 0, stream>>>(msg, (unsigned long long)N3 * 128);
    zero_f32<<<cdiv((unsigned)N3, 256), blk, 0, stream>>>(deg, (unsigned long long)N3);
    sage_scatter<<<cdiv((unsigned)E2, 8), blk, 0, stream>>>(h2, src2, dst2, E2, msg, deg);
    {
        unsigned tiles = cdiv((unsigned)N3, 16) * (64 / 16);
        sage_gemm<<<cdiv(tiles, 8), blk, 0, stream>>>(h2, msg, deg, Ws2, Wn2, b2, (float*)d_out, N3, 64, 0);
    }
}